// GraphAttentionLayer_61168924230410
// MI455X (gfx1250) — compile-verified
//
#include <hip/hip_runtime.h>
#include <hip/hip_bf16.h>

typedef float v2f __attribute__((ext_vector_type(2)));
typedef float v8f __attribute__((ext_vector_type(8)));

#define Bn 8
#define Cn 2048
#define Fn 64

// ---------------------------------------------------------------------------
// Kernel 1: Wh = X @ W^T  (flattened M = B*C = 16384, N = 64, K = 64)
// one wave (32 threads) computes a 16x64 tile with V_WMMA_F32_16X16X4_F32
// ---------------------------------------------------------------------------
__global__ __launch_bounds__(32) void gat_wh_kernel(const float* __restrict__ X,
                                                    const float* __restrict__ W,
                                                    float* __restrict__ Wh) {
  const int t    = blockIdx.x;          // 16-row tile index
  const int lane = threadIdx.x;
  const int half = lane >> 4;           // 0: K pair {k0,k0+1}, 1: {k0+2,k0+3}
  const int l15  = lane & 15;
  const int rowBase = t * 16;

  v8f acc[4] = {};                      // 4 N-tiles of 16 cols each

  const float* xrow = X + (size_t)(rowBase + l15) * Fn + half * 2;
#pragma unroll
  for (int k0 = 0; k0 < Fn; k0 += 4) {
    v2f a = *(const v2f*)(xrow + k0);   // A[m, k0+2*half .. +1]
    // B[k,n] = W[n,k]; lane supplies column n = nt*16 + l15
    const float* wcol = W + (size_t)l15 * Fn + k0 + half * 2;
#pragma unroll
    for (int nt = 0; nt < 4; ++nt) {
      v2f b = *(const v2f*)(wcol + nt * 16 * Fn);
      acc[nt] = __builtin_amdgcn_wmma_f32_16x16x4_f32(
          false, a, false, b, (short)0, acc[nt], false, false);
    }
  }

  // D layout: VGPR r -> row (r + 8*half), col = nt*16 + l15
#pragma unroll
  for (int nt = 0; nt < 4; ++nt) {
#pragma unroll
    for (int r = 0; r < 8; ++r) {
      Wh[(size_t)(rowBase + r + 8 * half) * Fn + nt * 16 + l15] = acc[nt][r];
    }
  }
}

// ---------------------------------------------------------------------------
// Kernel 2: s1 = Wh@a1, s2 = Wh@a2, s2max[b] = max_j s2[b,j]
// ---------------------------------------------------------------------------
__global__ __launch_bounds__(256) void gat_s_kernel(const float* __restrict__ Wh,
                                                    const float* __restrict__ a,
                                                    float* __restrict__ s1,
                                                    float* __restrict__ s2,
                                                    float* __restrict__ s2max) {
  const int b = blockIdx.x;
  __shared__ float red[256];
  float localmax = -3.4e38f;
  for (int c = threadIdx.x; c < Cn; c += 256) {
    const float* row = Wh + (size_t)(b * Cn + c) * Fn;
    float d1 = 0.f, d2 = 0.f;
#pragma unroll
    for (int n = 0; n < Fn; ++n) {
      float w = row[n];
      d1 += w * a[n];
      d2 += w * a[Fn + n];
    }
    s1[b * Cn + c] = d1;
    s2[b * Cn + c] = d2;
    localmax = fmaxf(localmax, d2);
  }
  red[threadIdx.x] = localmax;
  __syncthreads();
  for (int s = 128; s > 0; s >>= 1) {
    if (threadIdx.x < (unsigned)s) red[threadIdx.x] = fmaxf(red[threadIdx.x], red[threadIdx.x + s]);
    __syncthreads();
  }
  if (threadIdx.x == 0) s2max[b] = red[0];
}

// ---------------------------------------------------------------------------
// Kernel 3: per 16-row tile: softmax(leaky_relu(s1_i + s2_j)) -> attention,
// out = attention @ Wh (WMMA f32 16x16x4), elu(out).
// rowmax_i = lrelu(s1_i + max_j s2_j) exactly, since lrelu is monotone.
// ---------------------------------------------------------------------------
__global__ __launch_bounds__(32) void gat_attn_kernel(const float* __restrict__ Wh,
                                                      const float* __restrict__ s1g,
                                                      const float* __restrict__ s2g,
                                                      const float* __restrict__ s2maxg,
                                                      float* __restrict__ outElu,
                                                      float* __restrict__ outAttn) {
  const int t    = blockIdx.x;
  const int b    = t >> 7;              // 128 tiles per batch
  const int i0   = (t & 127) * 16;
  const int lane = threadIdx.x;
  const int half = lane >> 4;
  const int l15  = lane & 15;

  __shared__ __align__(16) float lds_p[16 * 68];   // stride 68 -> conflict-free frags

  const float  s2mx  = s2maxg[b];
  const float* s2row = s2g + b * Cn;

  // lane owns row l15 (duplicated in both halves)
  const float s1v = s1g[b * Cn + i0 + l15];
  float tmp = s1v + s2mx;
  const float rowmax = tmp > 0.f ? tmp : 0.2f * tmp;

  // ---- pass A: row sums of exp (lane: row l15, column parity = half) ----
  float accs = 0.f;
  for (int j = half; j < Cn; j += 2) {
    float x = s1v + s2row[j];
    x = x > 0.f ? x : 0.2f * x;
    accs += __expf(x - rowmax);
  }
  accs += __shfl_xor(accs, 16);
  const float inv = 1.0f / accs;

  v8f accD[4] = {};
  float* attnBase = outAttn + (size_t)(b * Cn + i0) * Cn;

  // ---- pass B: 64-column chunks ----
  for (int j0 = 0; j0 < Cn; j0 += 64) {
    // compute normalized p for the 16x64 chunk; write LDS + coalesced global
#pragma unroll
    for (int q = 0; q < 32; ++q) {
      const int row = q >> 1;
      const int col = ((q & 1) << 5) + lane;       // 32 consecutive cols per q
      const float s1r = __shfl(s1v, row);
      const float rmr = __shfl(rowmax, row);
      const float ivr = __shfl(inv, row);
      float x = s1r + s2row[j0 + col];
      x = x > 0.f ? x : 0.2f * x;
      const float p = __expf(x - rmr) * ivr;
      lds_p[row * 68 + col] = p;
      attnBase[(size_t)row * Cn + j0 + col] = p;   // 128B coalesced store
    }
    __syncthreads();

    const float* WhB = Wh + (size_t)(b * Cn + j0) * Fn;
#pragma unroll 4
    for (int k0 = 0; k0 < 64; k0 += 4) {
      v2f a = *(const v2f*)(&lds_p[l15 * 68 + k0 + half * 2]);  // A[m, k..k+1]
      const float* wb = WhB + (size_t)(k0 + half * 2) * Fn + l15;
#pragma unroll
      for (int nt = 0; nt < 4; ++nt) {
        v2f bf;
        bf.x = wb[nt * 16];            // B[k,   n] = Wh[j0+k0+2h,   n]
        bf.y = wb[nt * 16 + Fn];       // B[k+1, n] = Wh[j0+k0+2h+1, n]
        accD[nt] = __builtin_amdgcn_wmma_f32_16x16x4_f32(
            false, a, false, bf, (short)0, accD[nt], false, false);
      }
    }
    __syncthreads();
  }

  // ---- epilogue: ELU + store out tile ----
#pragma unroll
  for (int nt = 0; nt < 4; ++nt) {
#pragma unroll
    for (int r = 0; r < 8; ++r) {
      float v = accD[nt][r];
      float e = v > 0.f ? v : (__expf(v) - 1.0f);
      outElu[(size_t)(b * Cn + i0 + r + 8 * half) * Fn + nt * 16 + l15] = e;
    }
  }
}

// ---------------------------------------------------------------------------
extern "C" void kernel_launch(void* const* d_in, const int* in_sizes, int n_in,
                              void* d_out, int out_size, void* d_ws, size_t ws_size,
                              hipStream_t stream) {
  const float* X = (const float*)d_in[0];   // (8, 2048, 64)
  const float* W = (const float*)d_in[1];   // (64, 64)
  const float* a = (const float*)d_in[2];   // (128, 1)

  float* outElu  = (float*)d_out;                       // 8*2048*64
  float* outAttn = outElu + (size_t)Bn * Cn * Fn;       // 8*2048*2048

  float* Wh    = (float*)d_ws;                          // 8*2048*64 floats (4 MB)
  float* s1    = Wh + (size_t)Bn * Cn * Fn;
  float* s2    = s1 + (size_t)Bn * Cn;
  float* s2max = s2 + (size_t)Bn * Cn;

  gat_wh_kernel<<<(Bn * Cn) / 16, 32, 0, stream>>>(X, W, Wh);
  gat_s_kernel<<<Bn, 256, 0, stream>>>(Wh, a, s1, s2, s2max);
  gat_attn_kernel<<<(Bn * Cn) / 16, 32, 0, stream>>>(Wh, s1, s2, s2max, outElu, outAttn);
}